// GCN_Module_81398220194326
// MI455X (gfx1250) — compile-verified
//
#include <hip/hip_runtime.h>
#include <hip/hip_bf16.h>
#include <math.h>

typedef __attribute__((ext_vector_type(2))) float v2f;
typedef __attribute__((ext_vector_type(8))) float v8f;

#define DIN 128
#define LDA 132   // padded LDS row stride (132 % 64 == 4 -> bank-conflict-free)

// ---------------- utility fills ----------------
__global__ void k_fill(float* __restrict__ p, int n, float v) {
  int i = blockIdx.x * blockDim.x + threadIdx.x;
  if (i < n) p[i] = v;
}

// ---------------- degree / norm ----------------
__global__ void k_deg(const long long* __restrict__ ei, float* __restrict__ deg, int E) {
  int e = blockIdx.x * blockDim.x + threadIdx.x;
  if (e < E) {
    long long d = ei[(long long)E + e];
    __hip_atomic_fetch_add(&deg[d], 1.0f, __ATOMIC_RELAXED, __HIP_MEMORY_SCOPE_AGENT);
  }
}

__global__ void k_rsqrt(float* __restrict__ dinv, int n) {
  int i = blockIdx.x * blockDim.x + threadIdx.x;
  if (i < n) {
    float d = dinv[i];
    dinv[i] = (d > 0.0f) ? rsqrtf(d) : 0.0f;
  }
}

// ---------------- GEMM1: H = X @ W1 via fp32 WMMA ----------------
// block = 256 threads (8 waves). Block handles a 16-row slab of X;
// wave w computes the 16x16 output tile covering columns [16w, 16w+16).
__global__ __launch_bounds__(256) void k_gemm_wmma(const float* __restrict__ X,
                                                   const float* __restrict__ W,
                                                   float* __restrict__ H, int n) {
  __shared__ float ldsA[16 * LDA];
  const int m0 = blockIdx.x * 16;

  // cooperative, coalesced load of 16x128 A-slab (clamp OOB rows to 0)
  for (int idx = threadIdx.x; idx < 16 * DIN; idx += 256) {
    int r = idx >> 7, c = idx & 127;
    float v = 0.0f;
    if (m0 + r < n) v = X[(size_t)(m0 + r) * DIN + c];
    ldsA[r * LDA + c] = v;
  }
  __syncthreads();

  const int lane = threadIdx.x & 31;
  const int wv   = threadIdx.x >> 5;
  const int mh   = lane & 15;   // A row / B,D column within tile
  const int kh   = lane >> 4;   // K-half (A/B), M+8 half (D)
  const int n0   = wv * 16;

  v8f acc = {};
  const float* wp = W + n0 + mh;          // base of this lane's W1 column
  const float* ap = ldsA + mh * LDA;      // base of this lane's A row

  #pragma unroll 4
  for (int k0 = 0; k0 < DIN; k0 += 4) {
    const int ka = k0 + 2 * kh;
    v2f a, b;
    a.x = ap[ka];
    a.y = ap[ka + 1];
    b.x = wp[(size_t)ka * DIN];
    b.y = wp[(size_t)(ka + 1) * DIN];
    // (neg_a, A, neg_b, B, c_mod, C, reuse_a, reuse_b)
    acc = __builtin_amdgcn_wmma_f32_16x16x4_f32(false, a, false, b,
                                                (short)0, acc, false, false);
  }

  // D layout: VGPR v -> row m0+v (lanes 0-15) / m0+v+8 (lanes 16-31), col n0+mh
  const int rbase = m0 + 8 * kh;
  float* op = H + (size_t)rbase * DIN + (n0 + mh);
  if (m0 + 16 <= n) {
    // full tile: scalar-uniform branch, 8 stores off one base with imm offsets
    #pragma unroll
    for (int v = 0; v < 8; ++v) op[(size_t)v * DIN] = acc[v];
  } else {
    // grid-tail partial tile only
    #pragma unroll
    for (int v = 0; v < 8; ++v) {
      if (rbase + v < n) op[(size_t)v * DIN] = acc[v];
    }
  }
}

// ---------------- aggregation layer 1: wave32 per edge ----------------
__global__ __launch_bounds__(256) void k_agg1(const long long* __restrict__ ei,
                                              const float* __restrict__ dinv,
                                              const float* __restrict__ H,
                                              float* __restrict__ A1, int E) {
  const int wv   = threadIdx.x >> 5;
  const int lane = threadIdx.x & 31;
  const int e    = blockIdx.x * 8 + wv;
  if (e >= E) return;
  const long long s = ei[e];
  const long long d = ei[(long long)E + e];
  const float c = dinv[s] * dinv[d];
  const float4 v = *(const float4*)(H + (size_t)s * DIN + lane * 4);
  float* o = A1 + (size_t)d * DIN + lane * 4;
  __hip_atomic_fetch_add(o + 0, v.x * c, __ATOMIC_RELAXED, __HIP_MEMORY_SCOPE_AGENT);
  __hip_atomic_fetch_add(o + 1, v.y * c, __ATOMIC_RELAXED, __HIP_MEMORY_SCOPE_AGENT);
  __hip_atomic_fetch_add(o + 2, v.z * c, __ATOMIC_RELAXED, __HIP_MEMORY_SCOPE_AGENT);
  __hip_atomic_fetch_add(o + 3, v.w * c, __ATOMIC_RELAXED, __HIP_MEMORY_SCOPE_AGENT);
}

// finalize layer 1: add self-loop + bias, ReLU (in place into A1)
__global__ void k_fin1(const float* __restrict__ H, float* __restrict__ A1,
                       const float* __restrict__ dinv, const float* __restrict__ b1, int n) {
  int t = blockIdx.x * blockDim.x + threadIdx.x;
  if (t >= n * DIN) return;
  int i = t >> 7, j = t & 127;
  float di = dinv[i];
  float v = A1[t] + H[t] * di * di + b1[j];
  A1[t] = v > 0.0f ? v : 0.0f;
}

// ---------------- GEMM2: H2 = relu(A1) @ W2 (128 -> 2), wave per node ----------------
__global__ __launch_bounds__(256) void k_gemm2(const float* __restrict__ HR,
                                               const float* __restrict__ W2,
                                               float* __restrict__ H2, int n) {
  const int wv   = threadIdx.x >> 5;
  const int lane = threadIdx.x & 31;
  const int i    = blockIdx.x * 8 + wv;
  if (i >= n) return;
  const float4 h   = *(const float4*)(HR + (size_t)i * DIN + lane * 4);
  const float4 w01 = *(const float4*)(W2 + lane * 8);      // rows 4l, 4l+1 (2 cols each)
  const float4 w23 = *(const float4*)(W2 + lane * 8 + 4);  // rows 4l+2, 4l+3
  float s0 = h.x * w01.x + h.y * w01.z + h.z * w23.x + h.w * w23.z;
  float s1 = h.x * w01.y + h.y * w01.w + h.z * w23.y + h.w * w23.w;
  #pragma unroll
  for (int m = 16; m >= 1; m >>= 1) {
    s0 += __shfl_xor(s0, m, 32);
    s1 += __shfl_xor(s1, m, 32);
  }
  if (lane == 0) {
    H2[(size_t)i * 2 + 0] = s0;
    H2[(size_t)i * 2 + 1] = s1;
  }
}

// ---------------- aggregation layer 2: thread per edge ----------------
__global__ void k_agg2(const long long* __restrict__ ei, const float* __restrict__ dinv,
                       const float* __restrict__ H2, float* __restrict__ O, int E) {
  int e = blockIdx.x * blockDim.x + threadIdx.x;
  if (e >= E) return;
  const long long s = ei[e];
  const long long d = ei[(long long)E + e];
  const float c = dinv[s] * dinv[d];
  const float2 v = *(const float2*)(H2 + (size_t)s * 2);
  __hip_atomic_fetch_add(O + (size_t)d * 2 + 0, v.x * c, __ATOMIC_RELAXED, __HIP_MEMORY_SCOPE_AGENT);
  __hip_atomic_fetch_add(O + (size_t)d * 2 + 1, v.y * c, __ATOMIC_RELAXED, __HIP_MEMORY_SCOPE_AGENT);
}

// finalize layer 2: self-loop + bias + sigmoid
__global__ void k_fin2(const float* __restrict__ H2, float* __restrict__ O,
                       const float* __restrict__ dinv, const float* __restrict__ b2, int n) {
  int t = blockIdx.x * blockDim.x + threadIdx.x;
  if (t >= n * 2) return;
  int i = t >> 1, c = t & 1;
  float di = dinv[i];
  float v = O[t] + H2[t] * di * di + b2[c];
  O[t] = 1.0f / (1.0f + __expf(-v));
}

extern "C" void kernel_launch(void* const* d_in, const int* in_sizes, int n_in,
                              void* d_out, int out_size, void* d_ws, size_t ws_size,
                              hipStream_t stream) {
  (void)n_in; (void)out_size; (void)ws_size;
  const float*     x  = (const float*)d_in[0];
  const long long* ei = (const long long*)d_in[1];
  const float*     W1 = (const float*)d_in[2];
  const float*     b1 = (const float*)d_in[3];
  const float*     W2 = (const float*)d_in[4];
  const float*     b2 = (const float*)d_in[5];
  float* out = (float*)d_out;

  const int n = in_sizes[0] / DIN;
  const int E = in_sizes[1] / 2;

  // workspace layout (256B aligned)
  char* ws = (char*)d_ws;
  size_t off = 0;
  float* dinv = (float*)(ws + off); off += (((size_t)n * 4) + 255) & ~(size_t)255;
  float* H    = (float*)(ws + off); off += (((size_t)n * DIN * 4) + 255) & ~(size_t)255;
  float* A1   = (float*)(ws + off); off += (((size_t)n * DIN * 4) + 255) & ~(size_t)255;
  float* H2   = (float*)(ws + off);

  const int T = 256;
  // init (deg starts at 1.0 for the self-loop)
  k_fill<<<(n + T - 1) / T, T, 0, stream>>>(dinv, n, 1.0f);
  k_fill<<<(n * DIN + T - 1) / T, T, 0, stream>>>(A1, n * DIN, 0.0f);
  k_fill<<<(n * 2 + T - 1) / T, T, 0, stream>>>(out, n * 2, 0.0f);

  // symmetric normalization factors
  k_deg<<<(E + T - 1) / T, T, 0, stream>>>(ei, dinv, E);
  k_rsqrt<<<(n + T - 1) / T, T, 0, stream>>>(dinv, n);

  // layer 1
  k_gemm_wmma<<<(n + 15) / 16, T, 0, stream>>>(x, W1, H, n);
  k_agg1<<<(E + 7) / 8, T, 0, stream>>>(ei, dinv, H, A1, E);
  k_fin1<<<(n * DIN + T - 1) / T, T, 0, stream>>>(H, A1, dinv, b1, n);

  // layer 2
  k_gemm2<<<(n + 7) / 8, T, 0, stream>>>(A1, W2, H2, n);
  k_agg2<<<(E + T - 1) / T, T, 0, stream>>>(ei, dinv, H2, out, E);
  k_fin2<<<(n * 2 + T - 1) / T, T, 0, stream>>>(H2, out, dinv, b2, n);
}